// TemporalPatternAttentionMechanism_54013508715051
// MI455X (gfx1250) — compile-verified
//
#include <hip/hip_runtime.h>
#include <math.h>

// Problem constants (from reference): B=4096, L=168, A=128, FN=32, FS=1
#define B_N   4096
#define L_N   168
#define A_N   128
#define FN_N  32
#define TA_N  256   // 2*A
#define CIN_N 288   // 2*A + FN

typedef float v2f __attribute__((ext_vector_type(2)));
typedef float v8f __attribute__((ext_vector_type(8)));

static __device__ __forceinline__ v8f wmma_f32(v2f a, v2f b, v8f c) {
  // D = A(16x4 f32) * B(4x16 f32) + C(16x16 f32)
  // 8 args: (neg_a, A, neg_b, B, c_mod, C, reuse_a, reuse_b)
  return __builtin_amdgcn_wmma_f32_16x16x4_f32(false, a, false, b, (short)0, c,
                                               false, false);
}

// ---------------------------------------------------------------------------
// K1: w = query(4096x256) @ W_q(256x32)  -> ws_w (4096x32)
// One block per 128 batch rows. 128 threads = 4 waves; wave owns M-tiles
// {2w,2w+1} x N-tiles {0,1}.
// ---------------------------------------------------------------------------
__global__ void __launch_bounds__(128)
k_gemm_wq(const float* __restrict__ query, const float* __restrict__ Wq,
          float* __restrict__ w_out) {
  __shared__ __align__(16) float Qs[32][132]; // [k][row], padded stride
  __shared__ __align__(16) float Ks[32][36];  // [k][n], padded stride
  const int tid  = threadIdx.x;
  const int wave = tid >> 5;
  const int lane = tid & 31;
  const int half = lane >> 4;
  const int sub  = lane & 15;
  const int row0 = blockIdx.x * 128;

  const v8f vzero = {0.f,0.f,0.f,0.f,0.f,0.f,0.f,0.f};
  v8f acc[2][2];
  acc[0][0] = vzero; acc[0][1] = vzero; acc[1][0] = vzero; acc[1][1] = vzero;

  for (int kc = 0; kc < TA_N; kc += 32) {
    // Stage query slice 128 rows x 32 cols, transposed into Qs[k][row].
    for (int p = 0; p < 8; ++p) {
      int idx = tid + p * 128;          // 0..1023
      int r = idx >> 3;                 // 0..127
      int c = (idx & 7) << 2;           // 0..28
      float4 v = *(const float4*)(query + (size_t)(row0 + r) * TA_N + kc + c);
      Qs[c + 0][r] = v.x; Qs[c + 1][r] = v.y;
      Qs[c + 2][r] = v.z; Qs[c + 3][r] = v.w;
    }
    // Stage W_q slice 32x32.
    for (int p = 0; p < 2; ++p) {
      int idx = tid + p * 128;          // 0..255
      int r = idx >> 3;                 // 0..31
      int c = (idx & 7) << 2;           // 0..28
      *(float4*)&Ks[r][c] = *(const float4*)(Wq + (size_t)(kc + r) * FN_N + c);
    }
    __syncthreads();
    for (int ks = 0; ks < 32; ks += 4) {
      int klo = ks + 2 * half;
      v2f af[2], bf[2];
      for (int i = 0; i < 2; ++i) {
        int f = (2 * wave + i) * 16 + sub;
        af[i].x = Qs[klo][f]; af[i].y = Qs[klo + 1][f];
      }
      for (int j = 0; j < 2; ++j) {
        int n = j * 16 + sub;
        bf[j].x = Ks[klo][n]; bf[j].y = Ks[klo + 1][n];
      }
      acc[0][0] = wmma_f32(af[0], bf[0], acc[0][0]);
      acc[0][1] = wmma_f32(af[0], bf[1], acc[0][1]);
      acc[1][0] = wmma_f32(af[1], bf[0], acc[1][0]);
      acc[1][1] = wmma_f32(af[1], bf[1], acc[1][1]);
    }
    __syncthreads();
  }
  for (int i = 0; i < 2; ++i)
    for (int j = 0; j < 2; ++j) {
      int n = j * 16 + sub;
      for (int r = 0; r < 8; ++r) {
        int m = row0 + (2 * wave + i) * 16 + 8 * half + r;
        w_out[(size_t)m * FN_N + n] = acc[i][j][r];
      }
    }
}

// ---------------------------------------------------------------------------
// K2: one block per batch row b (4096 blocks, 128 threads = 4 waves).
//   cv(128x32) = attn[b]^T(128x168) @ conv_k(168x32) + conv_b   (f32 WMMA)
//   fused: while streaming attn[b], write the shifted copy attn[b,1:,:] out
//   then: s = cv@w, a = sigmoid(s), d = a@cv  -> ws_d
// attn[b] is read from HBM exactly once.
// ---------------------------------------------------------------------------
__global__ void __launch_bounds__(128)
k_conv_gate(const float* __restrict__ attn, const float* __restrict__ convk,
            const float* __restrict__ convb, const float* __restrict__ w_in,
            float* __restrict__ d_vec, float* __restrict__ out_states) {
  __shared__ __align__(16) float As[8][132];   // [l][f], padded
  __shared__ __align__(16) float Ks[8][36];    // [l][n], padded
  __shared__ float cvs[128][33];               // [f][n], padded
  __shared__ float wv[FN_N];
  __shared__ float av[A_N];

  const int tid  = threadIdx.x;
  const int wave = tid >> 5;
  const int lane = tid & 31;
  const int half = lane >> 4;
  const int sub  = lane & 15;
  const int b    = blockIdx.x;

  const float* ab = attn + (size_t)b * L_N * A_N;
  float* os = out_states + (size_t)b * (L_N - 1) * A_N;

  if (tid < FN_N) wv[tid] = w_in[(size_t)b * FN_N + tid];

  const v8f vzero = {0.f,0.f,0.f,0.f,0.f,0.f,0.f,0.f};
  v8f acc[2][2];
  acc[0][0] = vzero; acc[0][1] = vzero; acc[1][0] = vzero; acc[1][1] = vzero;

  for (int it = 0; it < L_N / 8; ++it) {       // 21 chunks of 8 rows
    int lc = it * 8;
    // Load 8x128 attn chunk; also emit the shifted copy in the same pass.
    for (int p = 0; p < 2; ++p) {
      int idx = tid + p * 128;                 // 0..255
      int r = idx >> 5;                        // 0..7
      int c = (idx & 31) << 2;                 // 0..124
      float4 v = *(const float4*)(ab + (size_t)(lc + r) * A_N + c);
      *(float4*)&As[r][c] = v;
      int l = lc + r;
      if (l >= 1)
        *(float4*)(os + (size_t)(l - 1) * A_N + c) = v;
    }
    // Load 8x32 conv_k chunk.
    if (tid < 64) {
      int r = tid >> 3;
      int c = (tid & 7) << 2;
      *(float4*)&Ks[r][c] = *(const float4*)(convk + (size_t)(lc + r) * FN_N + c);
    }
    __syncthreads();
    for (int ks = 0; ks < 8; ks += 4) {
      int klo = ks + 2 * half;
      v2f af[2], bf[2];
      for (int i = 0; i < 2; ++i) {
        int f = (2 * wave + i) * 16 + sub;
        af[i].x = As[klo][f]; af[i].y = As[klo + 1][f];
      }
      for (int j = 0; j < 2; ++j) {
        int n = j * 16 + sub;
        bf[j].x = Ks[klo][n]; bf[j].y = Ks[klo + 1][n];
      }
      acc[0][0] = wmma_f32(af[0], bf[0], acc[0][0]);
      acc[0][1] = wmma_f32(af[0], bf[1], acc[0][1]);
      acc[1][0] = wmma_f32(af[1], bf[0], acc[1][0]);
      acc[1][1] = wmma_f32(af[1], bf[1], acc[1][1]);
    }
    __syncthreads();
  }

  // cv -> LDS with conv_b bias.
  for (int i = 0; i < 2; ++i)
    for (int j = 0; j < 2; ++j) {
      int n = j * 16 + sub;
      float cb = convb[n];
      for (int r = 0; r < 8; ++r) {
        int f = (2 * wave + i) * 16 + 8 * half + r;
        cvs[f][n] = acc[i][j][r] + cb;
      }
    }
  __syncthreads();

  // s[f] = cv[f,:]·w ; a = sigmoid(s)
  {
    int f = tid;                               // 0..127
    float s = 0.f;
    for (int n = 0; n < FN_N; ++n) s += cvs[f][n] * wv[n];
    av[f] = 1.f / (1.f + __expf(-s));
  }
  __syncthreads();

  // d[n] = a·cv[:,n]
  if (tid < FN_N) {
    float d = 0.f;
    for (int f = 0; f < A_N; ++f) d += av[f] * cvs[f][tid];
    d_vec[(size_t)b * FN_N + tid] = d;
  }
}

// ---------------------------------------------------------------------------
// K3: new_attns(4096x128) = [query | d](4096x288) @ W_out(288x128) + b_out
// One block per 128 rows, 256 threads = 8 waves; wave w owns M-tile w and
// all 8 N-tiles.  W_out is read only 32x total (stays hot in L2).
// ---------------------------------------------------------------------------
__global__ void __launch_bounds__(256)
k_gemm_out(const float* __restrict__ query, const float* __restrict__ d_vec,
           const float* __restrict__ Wout, const float* __restrict__ bout,
           float* __restrict__ out_attns) {
  __shared__ __align__(16) float Xs[32][132];  // [k][row]
  __shared__ __align__(16) float Ws[32][132];  // [k][n]
  const int tid  = threadIdx.x;
  const int wave = tid >> 5;
  const int lane = tid & 31;
  const int half = lane >> 4;
  const int sub  = lane & 15;
  const int row0 = blockIdx.x * 128;

  const v8f vzero = {0.f,0.f,0.f,0.f,0.f,0.f,0.f,0.f};
  v8f acc[8];
  for (int j = 0; j < 8; ++j) acc[j] = vzero;

  for (int kc = 0; kc < CIN_N; kc += 32) {     // 9 chunks (288 = 9*32)
    // Stage X slice 128x32, transposed. Source is query for kc<256, else d.
    for (int p = 0; p < 4; ++p) {
      int idx = tid + p * 256;                 // 0..1023
      int r = idx >> 3;                        // 0..127
      int c = (idx & 7) << 2;                  // 0..28
      float4 v;
      if (kc < TA_N)
        v = *(const float4*)(query + (size_t)(row0 + r) * TA_N + kc + c);
      else
        v = *(const float4*)(d_vec + (size_t)(row0 + r) * FN_N + (kc - TA_N) + c);
      Xs[c + 0][r] = v.x; Xs[c + 1][r] = v.y;
      Xs[c + 2][r] = v.z; Xs[c + 3][r] = v.w;
    }
    // Stage W_out slice 32x128.
    for (int p = 0; p < 4; ++p) {
      int idx = tid + p * 256;                 // 0..1023
      int r = idx >> 5;                        // 0..31
      int c = (idx & 31) << 2;                 // 0..124
      *(float4*)&Ws[r][c] = *(const float4*)(Wout + (size_t)(kc + r) * A_N + c);
    }
    __syncthreads();
    for (int ks = 0; ks < 32; ks += 4) {
      int klo = ks + 2 * half;
      v2f af;
      {
        int f = wave * 16 + sub;
        af.x = Xs[klo][f]; af.y = Xs[klo + 1][f];
      }
      for (int j = 0; j < 8; ++j) {
        v2f bf;
        int n = j * 16 + sub;
        bf.x = Ws[klo][n]; bf.y = Ws[klo + 1][n];
        acc[j] = wmma_f32(af, bf, acc[j]);
      }
    }
    __syncthreads();
  }

  for (int j = 0; j < 8; ++j) {
    int n = j * 16 + sub;
    float bb = bout[n];
    for (int r = 0; r < 8; ++r) {
      int m = row0 + wave * 16 + 8 * half + r;
      out_attns[(size_t)m * A_N + n] = acc[j][r] + bb;
    }
  }
}

// ---------------------------------------------------------------------------
extern "C" void kernel_launch(void* const* d_in, const int* in_sizes, int n_in,
                              void* d_out, int out_size, void* d_ws, size_t ws_size,
                              hipStream_t stream) {
  (void)in_sizes; (void)n_in; (void)out_size; (void)ws_size;
  const float* query = (const float*)d_in[0];  // (B, 2A)
  const float* attn  = (const float*)d_in[1];  // (B, L, A)
  const float* Wq    = (const float*)d_in[2];  // (2A, FN)
  const float* convk = (const float*)d_in[3];  // (L, 1, 1, FN) == (L, FN)
  const float* convb = (const float*)d_in[4];  // (FN,)
  const float* Wout  = (const float*)d_in[5];  // (2A+FN, A)
  const float* bout  = (const float*)d_in[6];  // (A,)

  float* out        = (float*)d_out;
  float* out_attns  = out;                               // (B, A)
  float* out_states = out + (size_t)B_N * A_N;           // (B, L-1, A)

  float* ws_w = (float*)d_ws;                            // (B, FN)
  float* ws_d = ws_w + (size_t)B_N * FN_N;               // (B, FN)

  k_gemm_wq  <<<B_N / 128, 128, 0, stream>>>(query, Wq, ws_w);
  k_conv_gate<<<B_N,       128, 0, stream>>>(attn, convk, convb, ws_w, ws_d,
                                             out_states);
  k_gemm_out <<<B_N / 128, 256, 0, stream>>>(query, ws_d, Wout, bout, out_attns);
}